// GCN_12017318494615
// MI455X (gfx1250) — compile-verified
//
#include <hip/hip_runtime.h>

typedef __attribute__((ext_vector_type(2))) float v2f;
typedef __attribute__((ext_vector_type(8))) float v8f;

// ---------------------------------------------------------------- utilities

__global__ void fill_f32(float* __restrict__ p, long long n, float v) {
  long long i = (long long)blockIdx.x * blockDim.x + threadIdx.x;
  if (i < n) p[i] = v;
}

__global__ void deg_accum(const int* __restrict__ dst, float* __restrict__ deg, int E) {
  int e = blockIdx.x * blockDim.x + threadIdx.x;
  if (e < E) atomicAdd(&deg[dst[e]], 1.0f);
}

__global__ void deg_rsqrt(const float* __restrict__ deg, float* __restrict__ dinv, int n) {
  int i = blockIdx.x * blockDim.x + threadIdx.x;
  if (i < n) {
    float d = deg[i];
    dinv[i] = d > 0.f ? rsqrtf(d) : 0.f;
  }
}

// ------------------------------------- fp32 WMMA GEMM with LDS-staged A strip
// C[M,NC] = A[M,K] * B[K,NC], row-major fp32. One block per 16-row strip of C,
// one wave per 16-column tile (blockDim.x = 32*NC/16). The 16xTK A strip is
// cooperatively loaded (coalesced float4) into LDS and shared by all waves.
// LDS row stride = TK+4 floats so lanes 0-15 hit distinct banks on ds_load_b64.
//
// WMMA fragments (V_WMMA_F32_16X16X4_F32):
//  A (16x4, 2 VGPR): lanes 0-15 -> M=lane, K=k+0,k+1 ; lanes 16-31 -> K=k+2,k+3
//  B (4x16, 2 VGPR): lanes 0-15 -> N=lane, K=k+0,k+1 ; lanes 16-31 -> K=k+2,k+3
//  C/D (16x16, 8 VGPR): VGPR g -> row g + 8*(lane>=16), col = lane&15
template <int TK>
__global__ void gemm_wmma_f32_lds(const float* __restrict__ A,
                                  const float* __restrict__ B,
                                  float* __restrict__ C,
                                  int M, int K, int NC) {
  constexpr int LSTR = TK + 4;                 // padded row stride (floats)
  __shared__ float As[16 * LSTR];

  const int tid   = threadIdx.x;
  const int lane  = tid & 31;
  const int wave  = tid >> 5;                  // column tile within the strip
  const int row0  = blockIdx.x << 4;
  const int col0  = wave << 4;
  const int mn    = lane & 15;
  const int half  = lane >> 4;                 // 0 or 1
  const int half2 = half << 1;                 // K sub-offset 0 or 2
  const int nthr  = blockDim.x;

  const float* bp = B + (size_t)half2 * NC + col0 + mn;

  v8f acc = {};
  for (int k0 = 0; k0 < K; k0 += TK) {
    // cooperative coalesced stage of A[row0..row0+15, k0..k0+TK) into LDS
    for (int t = tid * 4; t < 16 * TK; t += nthr * 4) {
      const int r = t / TK;
      const int c = t % TK;
      const float* gp = A + (size_t)(row0 + r) * K + k0 + c;
      __builtin_prefetch(gp + TK, 0, 1);       // global_prefetch_b8: next chunk
      *(float4*)(&As[r * LSTR + c]) = *(const float4*)gp;
    }
    __syncthreads();

#pragma unroll
    for (int kk = 0; kk < TK; kk += 4) {
      v2f a, b;
      a.x = As[mn * LSTR + kk + half2];        // ds_load_b64, bank-conflict free
      a.y = As[mn * LSTR + kk + half2 + 1];
      b.x = bp[(size_t)(k0 + kk) * NC];        // coalesced 64B per half-wave
      b.y = bp[(size_t)(k0 + kk + 1) * NC];
      acc = __builtin_amdgcn_wmma_f32_16x16x4_f32(
          /*neg_a=*/false, a, /*neg_b=*/false, b,
          /*c_mod=*/(short)0, acc, /*reuse_a=*/false, /*reuse_b=*/false);
    }
    __syncthreads();
  }

  float* cp = C + (size_t)(row0 + (half << 3)) * NC + col0 + mn;
#pragma unroll
  for (int g = 0; g < 8; ++g) cp[(size_t)g * NC] = acc[g];
}

// ------------------- edge scatter: z[d, c:c+4] += h[s, c:c+4] * norm (float4)
template <int CH>
__global__ void scatter_edges4(const int* __restrict__ src, const int* __restrict__ dst,
                               const float* __restrict__ dinv,
                               const float* __restrict__ h, float* __restrict__ z,
                               int E) {
  constexpr int CV = CH / 4;
  long long idx = (long long)blockIdx.x * blockDim.x + threadIdx.x;
  if (idx >= (long long)E * CV) return;
  const int e  = (int)(idx / CV);
  const int c4 = (int)(idx % CV) * 4;
  const int s = src[e], d = dst[e];
  const float norm = dinv[s] * dinv[d];
  const float4 hv = *(const float4*)(h + (size_t)s * CH + c4);
  float* zp = z + (size_t)d * CH + c4;
  atomicAdd(zp + 0, hv.x * norm);              // global_atomic_add_f32
  atomicAdd(zp + 1, hv.y * norm);
  atomicAdd(zp + 2, hv.z * norm);
  atomicAdd(zp + 3, hv.w * norm);
}

// ------------- self-loop contribution + bias (+ optional ReLU), float4, in-place
template <int CH, bool RELU>
__global__ void self_bias4(const float* __restrict__ h, const float* __restrict__ dinv,
                           const float* __restrict__ bias, float* __restrict__ z,
                           int n) {
  constexpr int CV = CH / 4;
  long long idx = (long long)blockIdx.x * blockDim.x + threadIdx.x;
  if (idx >= (long long)n * CV) return;
  const int i  = (int)(idx / CV);
  const int c4 = (int)(idx % CV) * 4;
  const float dv = dinv[i];
  const float s = dv * dv;
  const float4 hv = *(const float4*)(h + (size_t)i * CH + c4);
  const float4 bv = *(const float4*)(bias + c4);
  float4* zp = (float4*)(z + (size_t)i * CH + c4);
  float4 zv = *zp;
  float4 r;
  r.x = zv.x + hv.x * s + bv.x;
  r.y = zv.y + hv.y * s + bv.y;
  r.z = zv.z + hv.z * s + bv.z;
  r.w = zv.w + hv.w * s + bv.w;
  if (RELU) {
    r.x = fmaxf(r.x, 0.f); r.y = fmaxf(r.y, 0.f);
    r.z = fmaxf(r.z, 0.f); r.w = fmaxf(r.w, 0.f);
  }
  *zp = r;
}

// --------------------------------------------- edge label dot products (64 ch)
__global__ void edge_dot(const int* __restrict__ es, const int* __restrict__ ed,
                         const float* __restrict__ z, float* __restrict__ out, int L) {
  int j = blockIdx.x * blockDim.x + threadIdx.x;
  if (j >= L) return;
  const float4* a = (const float4*)(z + (size_t)es[j] * 64);
  const float4* b = (const float4*)(z + (size_t)ed[j] * 64);
  float sum = 0.f;
#pragma unroll
  for (int t = 0; t < 16; ++t) {
    float4 x = a[t], y = b[t];
    sum += x.x * y.x + x.y * y.y + x.z * y.z + x.w * y.w;
  }
  out[j] = sum;
}

// ----------------------------------------------------------------- launcher

static inline int cdiv_ll(long long a, long long b) { return (int)((a + b - 1) / b); }

extern "C" void kernel_launch(void* const* d_in, const int* in_sizes, int n_in,
                              void* d_out, int out_size, void* d_ws, size_t ws_size,
                              hipStream_t stream) {
  const float* x   = (const float*)d_in[0];
  const int*   ei  = (const int*)d_in[1];
  const int*   eli = (const int*)d_in[2];
  const float* W1  = (const float*)d_in[3];
  const float* b1  = (const float*)d_in[4];
  const float* W2  = (const float*)d_in[5];
  const float* b2  = (const float*)d_in[6];
  float* out = (float*)d_out;

  const int IN = 512, HID = 128, OUT = 64;
  const int n = in_sizes[0] / IN;     // 50000, divisible by 16
  const int E = in_sizes[1] / 2;      // 800000
  const int L = in_sizes[2] / 2;      // 100000
  const int* src = ei;
  const int* dst = ei + E;
  const int* es  = eli;
  const int* ed  = eli + L;

  // workspace layout (floats): deg[n] dinv[n] h1[n*128] z1[n*128] h2[n*64] z2[n*64]
  float* ws   = (float*)d_ws;
  float* deg  = ws;  ws += n;
  float* dinv = ws;  ws += n;
  float* h1   = ws;  ws += (size_t)n * HID;
  float* z1   = ws;  ws += (size_t)n * HID;
  float* h2   = ws;  ws += (size_t)n * OUT;
  float* z2   = ws;  ws += (size_t)n * OUT;

  const int B = 256;

  // init every call: deg starts at 1.0 (self loop), accumulators at 0
  fill_f32<<<cdiv_ll(n, B), B, 0, stream>>>(deg, n, 1.0f);
  fill_f32<<<cdiv_ll((long long)n * HID, B), B, 0, stream>>>(z1, (long long)n * HID, 0.f);
  fill_f32<<<cdiv_ll((long long)n * OUT, B), B, 0, stream>>>(z2, (long long)n * OUT, 0.f);

  deg_accum<<<cdiv_ll(E, B), B, 0, stream>>>(dst, deg, E);
  deg_rsqrt<<<cdiv_ll(n, B), B, 0, stream>>>(deg, dinv, n);

  // layer 1: h1 = x @ W1 ; z1 = relu(scatter(h1) + selfloop + b1)
  gemm_wmma_f32_lds<64><<<n / 16, 32 * (HID / 16), 0, stream>>>(x, W1, h1, n, IN, HID);
  scatter_edges4<128><<<cdiv_ll((long long)E * (HID / 4), B), B, 0, stream>>>(src, dst, dinv, h1, z1, E);
  self_bias4<128, true><<<cdiv_ll((long long)n * (HID / 4), B), B, 0, stream>>>(h1, dinv, b1, z1, n);

  // layer 2: h2 = z1 @ W2 ; z2 = scatter(h2) + selfloop + b2
  gemm_wmma_f32_lds<64><<<n / 16, 32 * (OUT / 16), 0, stream>>>(z1, W2, h2, n, HID, OUT);
  scatter_edges4<64><<<cdiv_ll((long long)E * (OUT / 4), B), B, 0, stream>>>(src, dst, dinv, h2, z2, E);
  self_bias4<64, false><<<cdiv_ll((long long)n * (OUT / 4), B), B, 0, stream>>>(h2, dinv, b2, z2, n);

  // edge label dot products
  edge_dot<<<cdiv_ll(L, B), B, 0, stream>>>(es, ed, z2, out, L);
}